// CantorMultiheadFusionV2_13391708028928
// MI455X (gfx1250) — compile-verified
//
#include <hip/hip_runtime.h>
#include <hip/hip_bf16.h>

#define S_LEN   2048
#define D_DIM   512
#define B_SZ    2
#define KN      64
#define M_ROWS  (B_SZ * S_LEN)   // 4096

typedef __attribute__((ext_vector_type(2))) float v2f;
typedef __attribute__((ext_vector_type(8))) float v8f;

// ---------------------------------------------------------------------------
// Kernel 1: Devil's-staircase Cantor measure, cm[S]
// ---------------------------------------------------------------------------
__global__ __launch_bounds__(256) void cantor_kernel(float* __restrict__ cm) {
    int i = blockIdx.x * blockDim.x + threadIdx.x;
    if (i >= S_LEN) return;
    float pos = ((float)i + 0.5f) / (float)S_LEN;
    pos = fminf(fmaxf(pos, 1e-10f), 1.0f - 1e-10f);
    float acc = 0.0f;
    float scale = 1.0f, lw = 1.0f;
#pragma unroll
    for (int k = 1; k <= 5; ++k) {
        scale *= 3.0f;
        lw *= 0.5f;
        float y = fmodf(pos * scale, 3.0f);
        float inv_tau = 1.0f / 0.25f;      // fp32(0.25 + 1e-10) == 0.25
        float t0 = -(y - 0.5f) * (y - 0.5f) * inv_tau;
        float t1 = -(y - 1.5f) * (y - 1.5f) * inv_tau;
        float t2 = -(y - 2.5f) * (y - 2.5f) * inv_tau;
        float mm = fmaxf(t0, fmaxf(t1, t2));
        float e0 = expf(t0 - mm), e1 = expf(t1 - mm), e2 = expf(t2 - mm);
        float inv = 1.0f / (e0 + e1 + e2);
        float bits = (e2 + 0.5f * e1) * inv;   // p2 + ALPHA*p1
        acc += bits * lw;
    }
    cm[i] = acc;
}

// ---------------------------------------------------------------------------
// Kernel 2: per-row top-64 by |cm_s - cm_j| (normalized by global range),
// tie-break on lower index (matches lax.top_k), then softmax(1/(d+eps)).
// One workgroup (256 threads) per row s.
// ---------------------------------------------------------------------------
__global__ __launch_bounds__(256) void topk_kernel(const float* __restrict__ cm,
                                                   int* __restrict__ routes,
                                                   float* __restrict__ w) {
    __shared__ float sd[S_LEN];
    __shared__ unsigned long long red[256];
    __shared__ float rmin[256], rmax[256];
    __shared__ float seld[KN];

    int s = blockIdx.x;
    int t = threadIdx.x;
    float cms = cm[s];

    float vals[8];
    float lmin = 1e30f, lmax = -1e30f;
#pragma unroll
    for (int i = 0; i < 8; ++i) {
        float c = cm[t + i * 256];
        vals[i] = c;
        lmin = fminf(lmin, c);
        lmax = fmaxf(lmax, c);
    }
    rmin[t] = lmin; rmax[t] = lmax;
    __syncthreads();
    for (int off = 128; off > 0; off >>= 1) {
        if (t < off) {
            rmin[t] = fminf(rmin[t], rmin[t + off]);
            rmax[t] = fmaxf(rmax[t], rmax[t + off]);
        }
        __syncthreads();
    }
    float dmax = (rmax[0] - rmin[0]) + 1e-10f;   // Dm.max() + 1e-10
    float inv_dmax = 1.0f / dmax;

#pragma unroll
    for (int i = 0; i < 8; ++i) {
        int j = t + i * 256;
        sd[j] = fabsf(cms - vals[i]) * inv_dmax;
    }
    __syncthreads();

    for (int k = 0; k < KN; ++k) {
        unsigned long long best = 0xFFFFFFFFFFFFFFFFull;
#pragma unroll
        for (int i = 0; i < 8; ++i) {
            int j = t + i * 256;
            unsigned long long key =
                (((unsigned long long)__float_as_uint(sd[j])) << 32) | (unsigned)j;
            best = (key < best) ? key : best;
        }
        red[t] = best;
        __syncthreads();
        for (int off = 128; off > 0; off >>= 1) {
            if (t < off) { if (red[t + off] < red[t]) red[t] = red[t + off]; }
            __syncthreads();
        }
        if (t == 0) {
            unsigned long long kb = red[0];
            int j = (int)(kb & 0xFFFFFFFFu);
            routes[s * KN + k] = j;
            seld[k] = __uint_as_float((unsigned)(kb >> 32));
            sd[j] = 1e30f;   // remove from candidate set
        }
        __syncthreads();
    }

    if (t == 0) {
        float lv[KN];
        float m = -1e30f;
#pragma unroll
        for (int k = 0; k < KN; ++k) {
            lv[k] = 1.0f / (seld[k] + 1e-8f);
            m = fmaxf(m, lv[k]);
        }
        float ssum = 0.0f;
#pragma unroll
        for (int k = 0; k < KN; ++k) { lv[k] = expf(lv[k] - m); ssum += lv[k]; }
        float inv = 1.0f / ssum;
#pragma unroll
        for (int k = 0; k < KN; ++k) w[s * KN + k] = lv[k] * inv;
    }
}

// ---------------------------------------------------------------------------
// Kernel 3/5: fp32 GEMM via V_WMMA_F32_16X16X4_F32.
//   C[M x 512] = A[M x 512] * Bm[512 x 512] + bias (+ resid)
// Block: 256 threads = 8 waves, 2(M) x 4(N) wave grid.
// Wave tile 32(M) x 64(N): each B fragment feeds two WMMAs (A-row reuse x2),
// cutting cache traffic per output from 160B to 96B. Block tile 64 x 256.
// Operand layouts per CDNA5 ISA 7.12.2:
//   A 16x4 : lane m = lane&15, VGPR0/1 = K {khalf, khalf+1}, khalf = (lane>=16)*2
//   B 4x16 : lane n = lane&15, VGPR0/1 = rows {khalf, khalf+1}
//   C 16x16: VGPR i -> row i + 8*(lane>=16), col = lane&15
// ---------------------------------------------------------------------------
__global__ __launch_bounds__(256) void gemm_f32_wmma_kernel(
        const float* __restrict__ A, const float* __restrict__ Bm,
        const float* __restrict__ bias, const float* __restrict__ resid,
        float* __restrict__ C) {
    int wave = threadIdx.x >> 5;
    int lane = threadIdx.x & 31;
    int m0 = blockIdx.x * 64 + (wave >> 2) * 32;
    int n0 = blockIdx.y * 256 + (wave & 3) * 64;
    int ml = lane & 15;
    int khalf = (lane >> 4) * 2;

    v8f c[2][4];
#pragma unroll
    for (int mi = 0; mi < 2; ++mi)
#pragma unroll
        for (int tn = 0; tn < 4; ++tn)
            c[mi][tn] = (v8f){0, 0, 0, 0, 0, 0, 0, 0};

    const float* arow0 = A + (size_t)(m0 + ml) * D_DIM;
    const float* arow1 = arow0 + (size_t)16 * D_DIM;

    for (int k0 = 0; k0 < D_DIM; k0 += 4) {
        if (k0 + 16 < D_DIM) {
            __builtin_prefetch(arow0 + k0 + 16 + khalf, 0, 1);  // global_prefetch_b8
            __builtin_prefetch(arow1 + k0 + 16 + khalf, 0, 1);
        }
        // 8-byte aligned: row base is 2KB aligned, k0+khalf even
        v2f a0 = *(const v2f*)(arow0 + k0 + khalf);
        v2f a1 = *(const v2f*)(arow1 + k0 + khalf);
#pragma unroll
        for (int tn = 0; tn < 4; ++tn) {
            int n = n0 + tn * 16 + ml;
            v2f b;
            b.x = Bm[(size_t)(k0 + khalf) * D_DIM + n];
            b.y = Bm[(size_t)(k0 + khalf + 1) * D_DIM + n];
            c[0][tn] = __builtin_amdgcn_wmma_f32_16x16x4_f32(
                false, a0, false, b, (short)0, c[0][tn], false, false);
            c[1][tn] = __builtin_amdgcn_wmma_f32_16x16x4_f32(
                false, a1, false, b, (short)0, c[1][tn], false, false);
        }
    }

#pragma unroll
    for (int mi = 0; mi < 2; ++mi) {
        int rowbase = m0 + mi * 16 + (lane >> 4) * 8;
#pragma unroll
        for (int tn = 0; tn < 4; ++tn) {
            int col = n0 + tn * 16 + ml;
            float bv = bias[col];
#pragma unroll
            for (int i = 0; i < 8; ++i) {
                size_t idx = (size_t)(rowbase + i) * D_DIM + col;
                float v = c[mi][tn][i] + bv;
                if (resid) v += resid[idx];
                C[idx] = v;
            }
        }
    }
}

// ---------------------------------------------------------------------------
// Kernel 4: gather-weighted fusion. Heads are irrelevant: routes are shared
// across heads, so it's a row gather on the [M x 512] matrix h.
// One block (128 threads, float4 each) per output row; h is L2-resident.
// ---------------------------------------------------------------------------
__global__ __launch_bounds__(128) void fuse_kernel(const float* __restrict__ h,
                                                   const int* __restrict__ routes,
                                                   const float* __restrict__ w,
                                                   float* __restrict__ f) {
    __shared__ int   rsh[KN];
    __shared__ float wsh[KN];
    int row = blockIdx.x;            // b*S + s
    int b = row >> 11;
    int s = row & (S_LEN - 1);
    int t = threadIdx.x;             // 0..127, float4 lanes over D=512
    if (t < KN) {
        rsh[t] = routes[s * KN + t];
        wsh[t] = w[s * KN + t];
    }
    __syncthreads();

    const float4* hb = (const float4*)(h + (size_t)b * S_LEN * D_DIM);
    float4 acc = {0.0f, 0.0f, 0.0f, 0.0f};
#pragma unroll 4
    for (int k = 0; k < KN; ++k) {
        float4 v = hb[(size_t)rsh[k] * (D_DIM / 4) + t];
        float wt = wsh[k];
        acc.x += wt * v.x;
        acc.y += wt * v.y;
        acc.z += wt * v.z;
        acc.w += wt * v.w;
    }
    ((float4*)f)[(size_t)row * (D_DIM / 4) + t] = acc;
}

// ---------------------------------------------------------------------------
// launch
// ---------------------------------------------------------------------------
extern "C" void kernel_launch(void* const* d_in, const int* in_sizes, int n_in,
                              void* d_out, int out_size, void* d_ws, size_t ws_size,
                              hipStream_t stream) {
    const float* x     = (const float*)d_in[0];
    const float* W_in  = (const float*)d_in[1];
    const float* b_in  = (const float*)d_in[2];
    const float* W_out = (const float*)d_in[3];
    const float* b_out = (const float*)d_in[4];
    float* out = (float*)d_out;

    char* ws = (char*)d_ws;
    size_t off = 0;
    float* cm     = (float*)(ws + off); off += (size_t)S_LEN * sizeof(float);            // 8 KB
    int*   routes = (int*)  (ws + off); off += (size_t)S_LEN * KN * sizeof(int);         // 512 KB
    float* w      = (float*)(ws + off); off += (size_t)S_LEN * KN * sizeof(float);       // 512 KB
    float* h      = (float*)(ws + off); off += (size_t)M_ROWS * D_DIM * sizeof(float);   // 8 MB
    float* f      = (float*)(ws + off); off += (size_t)M_ROWS * D_DIM * sizeof(float);   // 8 MB
    (void)ws_size; (void)in_sizes; (void)n_in; (void)out_size;

    cantor_kernel<<<(S_LEN + 255) / 256, 256, 0, stream>>>(cm);
    topk_kernel<<<S_LEN, 256, 0, stream>>>(cm, routes, w);

    dim3 ggrid(M_ROWS / 64, D_DIM / 256);   // 64 x 2
    gemm_f32_wmma_kernel<<<ggrid, 256, 0, stream>>>(x, W_in, b_in, nullptr, h);
    fuse_kernel<<<M_ROWS, 128, 0, stream>>>(h, routes, w, f);
    gemm_f32_wmma_kernel<<<ggrid, 256, 0, stream>>>(f, W_out, b_out, x, out);
}